// GraphBlock_4037269258334
// MI455X (gfx1250) — compile-verified
//
#include <hip/hip_runtime.h>

typedef __attribute__((ext_vector_type(2))) float v2f;
typedef __attribute__((ext_vector_type(8))) float v8f;

#define NCC_EPS 1e-8f
#define NEG_SLOPE 0.2f

// ---------------------------------------------------------------------------
// Tiled transpose with optional bias: dst[cc][rr] = src[rr][cc] + bias[cc]
// src: [R][C] row-major, dst: [C][R]. R, C multiples of 32. grid = (R/32, C/32)
// ---------------------------------------------------------------------------
__global__ __launch_bounds__(256) void k_transpose_bias(
    const float* __restrict__ src, float* __restrict__ dst,
    const float* __restrict__ bias, int R, int C)
{
    __shared__ float tile[32][33];
    const int r0 = blockIdx.x * 32, c0 = blockIdx.y * 32;
    const int tx = threadIdx.x & 31, ty = threadIdx.x >> 5;   // 32 x 8
    for (int r = ty; r < 32; r += 8)
        tile[r][tx] = src[(size_t)(r0 + r) * C + (c0 + tx)];
    __syncthreads();
    for (int r = ty; r < 32; r += 8) {
        float b = bias ? bias[c0 + r] : 0.0f;
        dst[(size_t)(c0 + r) * R + (r0 + tx)] = tile[tx][r] + b;
    }
}

// ---------------------------------------------------------------------------
// NT GEMM, fp32 WMMA: C[i][j] = sum_k A[i][k] * B[j][k]
// A: [M][Kd], B: [N][Kd], C: [M][N]; M,N % 64 == 0, Kd % 16 == 0.
// 256 threads = 8 wave32; block tile 64x64; wave (mw,nw) owns 32x16 via two
// V_WMMA_F32_16X16X4_F32 accumulators. LDS rows padded to 20 floats:
// row starts 16B-aligned (float4 stores) and the 16-lane float2 fragment
// reads cover all 64 banks exactly once (20*r mod 64 hits every multiple of 4).
// grid = (N/64, M/64)
// ---------------------------------------------------------------------------
__global__ __launch_bounds__(256) void k_gemm_nt_wmma(
    const float* __restrict__ A, const float* __restrict__ B,
    float* __restrict__ C, int M, int N, int Kd)
{
    __shared__ float As[64][20];
    __shared__ float Bs[64][20];

    const int t    = threadIdx.x;
    const int lane = t & 31;
    const int wave = t >> 5;
    const int mw   = wave & 1;          // 0..1 : 32-row strip
    const int nw   = wave >> 1;         // 0..3 : 16-col strip
    const int biM  = blockIdx.y * 64;
    const int bjN  = blockIdx.x * 64;

    // cooperative loaders: 256 threads x float4 = 64x16 tile
    const int lr = t >> 2;              // 0..63
    const int lc = (t & 3) << 2;        // 0,4,8,12

    // fragment addressing (ISA 7.12.2): lane&15 = row, VGPR pair at k + 2*(lane>>4)
    const int fr = lane & 15;
    const int kh = (lane >> 4) << 1;    // 0 or 2

    v8f acc0 = {};
    v8f acc1 = {};

    const float* Ar = A + (size_t)(biM + lr) * Kd + lc;
    const float* Br = B + (size_t)(bjN + lr) * Kd + lc;

    for (int k0 = 0; k0 < Kd; k0 += 16) {
        float4 a4 = *(const float4*)(Ar + k0);
        float4 b4 = *(const float4*)(Br + k0);
        __syncthreads();
        *(float4*)&As[lr][lc] = a4;
        *(float4*)&Bs[lr][lc] = b4;
        __syncthreads();
#pragma unroll
        for (int ks = 0; ks < 16; ks += 4) {
            v2f af0 = *(const v2f*)&As[mw * 32 + fr][ks + kh];
            v2f af1 = *(const v2f*)&As[mw * 32 + 16 + fr][ks + kh];
            v2f bf  = *(const v2f*)&Bs[nw * 16 + fr][ks + kh];
            acc0 = __builtin_amdgcn_wmma_f32_16x16x4_f32(
                false, af0, false, bf, (short)0, acc0, false, false);
            acc1 = __builtin_amdgcn_wmma_f32_16x16x4_f32(
                false, af1, false, bf, (short)0, acc1, false, false);
        }
    }

    // C layout: VGPR r -> M=r (lanes 0-15) / M=r+8 (lanes 16-31), N = lane&15
    const int l16 = lane & 15;
    const int lh8 = (lane >> 4) << 3;
    const size_t col = (size_t)(bjN + nw * 16 + l16);
#pragma unroll
    for (int r = 0; r < 8; ++r) {
        C[(size_t)(biM + mw * 32 + lh8 + r) * N + col]      = acc0[r];
        C[(size_t)(biM + mw * 32 + 16 + lh8 + r) * N + col] = acc1[r];
    }
}

// ---------------------------------------------------------------------------
// Row squared norms: norms[row] = sum_c X[row][c]^2. grid = rows
// ---------------------------------------------------------------------------
__global__ __launch_bounds__(256) void k_rownorm(
    const float* __restrict__ X, float* __restrict__ norms, int f)
{
    __shared__ float s[256];
    const int row = blockIdx.x, t = threadIdx.x;
    const float* xr = X + (size_t)row * f;
    float acc = 0.f;
    for (int c = t; c < f; c += 256) { float v = xr[c]; acc += v * v; }
    s[t] = acc; __syncthreads();
    for (int off = 128; off > 0; off >>= 1) {
        if (t < off) s[t] += s[t + off];
        __syncthreads();
    }
    if (t == 0) norms[row] = s[0];
}

// ---------------------------------------------------------------------------
// Dual per-row dot: el[row] = X[row].al ; er[row] = X[row].ar. grid = rows
// ---------------------------------------------------------------------------
__global__ __launch_bounds__(256) void k_elr(
    const float* __restrict__ X, const float* __restrict__ al,
    const float* __restrict__ ar, float* __restrict__ el,
    float* __restrict__ er, int f)
{
    __shared__ float s1[256], s2[256];
    const int row = blockIdx.x, t = threadIdx.x;
    const float* xr = X + (size_t)row * f;
    float a = 0.f, b = 0.f;
    for (int c = t; c < f; c += 256) { float v = xr[c]; a += v * al[c]; b += v * ar[c]; }
    s1[t] = a; s2[t] = b; __syncthreads();
    for (int off = 128; off > 0; off >>= 1) {
        if (t < off) { s1[t] += s1[t + off]; s2[t] += s2[t + off]; }
        __syncthreads();
    }
    if (t == 0) { el[row] = s1[0]; er[row] = s2[0]; }
}

// ---------------------------------------------------------------------------
// Top-5 over NCC-normalized row: D[j] = (xy[j]+eps)/(sqrt(yy*xx[j])+eps)+1e-5
// 5 argmax passes; ties prefer lowest index (matches jax.lax.top_k).
// grid = rows of xy
// ---------------------------------------------------------------------------
__global__ __launch_bounds__(256) void k_topk(
    const float* __restrict__ xy, const float* __restrict__ normY,
    const float* __restrict__ normX, int n, int* __restrict__ idx)
{
    __shared__ float sv[256];
    __shared__ int   si[256];
    const int row = blockIdx.x, t = threadIdx.x;
    const float yy = normY[row];
    const float* xr = xy + (size_t)row * n;
    int chosen[5];
    for (int kk = 0; kk < 5; ++kk) {
        float bv = -3.4e38f; int bidx = n;
        for (int j = t; j < n; j += 256) {
            bool skip = false;
            for (int q = 0; q < kk; ++q) skip = skip || (chosen[q] == j);
            if (skip) continue;
            float d = (xr[j] + NCC_EPS) / (sqrtf(yy * normX[j]) + NCC_EPS) + 1e-5f;
            if (d > bv || (d == bv && j < bidx)) { bv = d; bidx = j; }
        }
        sv[t] = bv; si[t] = bidx; __syncthreads();
        for (int off = 128; off > 0; off >>= 1) {
            if (t < off) {
                if (sv[t + off] > sv[t] ||
                    (sv[t + off] == sv[t] && si[t + off] < si[t])) {
                    sv[t] = sv[t + off]; si[t] = si[t + off];
                }
            }
            __syncthreads();
        }
        if (t == 0) idx[row * 5 + kk] = si[0];
        chosen[kk] = si[0];
        __syncthreads();   // before sv/si are overwritten next pass
    }
}

// ---------------------------------------------------------------------------
// Per-dst-node edge softmax (10 edges: 5 content nbrs, 5 style nbrs at +m)
// + weighted feature aggregation: agg[v] = sum_j alpha_j * ft[u_j]. grid = m
// ---------------------------------------------------------------------------
__global__ __launch_bounds__(256) void k_agg(
    const float* __restrict__ ft, const float* __restrict__ el,
    const float* __restrict__ er, const int* __restrict__ idx2,
    const int* __restrict__ idx1, float* __restrict__ agg, int m, int f)
{
    __shared__ int   us[10];
    __shared__ float alpha[10];
    const int v = blockIdx.x, t = threadIdx.x;
    if (t < 5)       us[t] = idx2[v * 5 + t];
    else if (t < 10) us[t] = idx1[v * 5 + (t - 5)] + m;
    __syncthreads();
    if (t == 0) {
        const float erv = er[v];
        float e[10]; float mx = -3.4e38f;
        for (int j = 0; j < 10; ++j) {
            float x = el[us[j]] + erv;
            x = (x >= 0.f) ? x : NEG_SLOPE * x;
            e[j] = x; mx = fmaxf(mx, x);
        }
        float s = 0.f;
        for (int j = 0; j < 10; ++j) { e[j] = expf(e[j] - mx); s += e[j]; }
        for (int j = 0; j < 10; ++j) alpha[j] = e[j] / s;
    }
    __syncthreads();
    for (int c = t; c < f; c += 256) {
        float s = 0.f;
#pragma unroll
        for (int j = 0; j < 10; ++j) s += alpha[j] * ft[(size_t)us[j] * f + c];
        agg[(size_t)v * f + c] = s;
    }
}

// ---------------------------------------------------------------------------
// h[node][c] = relu((node < m ? agg[node][c] : 0) + bias[c])
// ---------------------------------------------------------------------------
__global__ __launch_bounds__(256) void k_bias_relu(
    const float* __restrict__ agg, const float* __restrict__ bias,
    float* __restrict__ h, int m, int Nnodes, int f)
{
    size_t i = (size_t)blockIdx.x * 256 + threadIdx.x;
    size_t tot = (size_t)Nnodes * f;
    if (i >= tot) return;
    int node = (int)(i / f);
    int c    = (int)(i % f);
    float x = bias[c] + (node < m ? agg[i] : 0.f);
    h[i] = x > 0.f ? x : 0.f;
}

// ---------------------------------------------------------------------------
// Host driver. Problem sizes fixed by setup_inputs(): b=2, c=256, p=5, g=32.
// Workspace (floats, reused across the 2 batches):
//   feat [2048*6400] | ft [2048*6400] (KNN D matrix overlaps here) |
//   agg [1024*6400]  | norms[2048] el[2048] er[2048] | idx1[5120] idx2[5120]
//   total ~131 MB.
// ---------------------------------------------------------------------------
extern "C" void kernel_launch(void* const* d_in, const int* in_sizes, int n_in,
                              void* d_out, int out_size, void* d_ws, size_t ws_size,
                              hipStream_t stream)
{
    (void)in_sizes; (void)n_in; (void)out_size; (void)ws_size;
    const float* ys  = (const float*)d_in[0];
    const float* yc  = (const float*)d_in[1];
    const float* W1  = (const float*)d_in[2];
    const float* al1 = (const float*)d_in[3];
    const float* ar1 = (const float*)d_in[4];
    const float* b1  = (const float*)d_in[5];
    const float* W2  = (const float*)d_in[6];
    const float* al2 = (const float*)d_in[7];
    const float* ar2 = (const float*)d_in[8];
    const float* b2  = (const float*)d_in[9];
    float* out = (float*)d_out;

    const int f = 6400, m = 1024, n = 1024, NN = 2048;
    const size_t NF = (size_t)NN * f;
    const size_t MF = (size_t)m * f;

    float* feat  = (float*)d_ws;
    float* ft    = feat + NF;
    float* Dbuf  = ft;                 // KNN scratch overlaps ft
    float* agg   = ft + NF;
    float* norms = agg + MF;
    float* el    = norms + NN;
    float* er    = el + NN;
    int*   idx1  = (int*)(er + NN);
    int*   idx2  = idx1 + m * 5;

    for (int bi = 0; bi < 2; ++bi) {
        const float* ysb  = ys + (size_t)bi * n * f;
        const float* ycb  = yc + (size_t)bi * f * m;   // [c,p,p,g,g] == [f][m]
        float*       outb = out + (size_t)bi * (size_t)f * m;

        // feat = [feat_c ; feat_s]; feat_c is transpose of yc's [f][m] block
        k_transpose_bias<<<dim3(f / 32, m / 32), 256, 0, stream>>>(
            ycb, feat, nullptr, f, m);
        hipMemcpyAsync(feat + MF, ysb, (size_t)n * f * sizeof(float),
                       hipMemcpyDeviceToDevice, stream);
        k_rownorm<<<NN, 256, 0, stream>>>(feat, norms, f);

        // KNN1: content rows vs style rows
        k_gemm_nt_wmma<<<dim3(n / 64, m / 64), 256, 0, stream>>>(
            feat, feat + MF, Dbuf, m, n, f);
        k_topk<<<m, 256, 0, stream>>>(Dbuf, norms, norms + m, n, idx1);
        // KNN2: content vs content
        k_gemm_nt_wmma<<<dim3(m / 64, m / 64), 256, 0, stream>>>(
            feat, feat, Dbuf, m, m, f);
        k_topk<<<m, 256, 0, stream>>>(Dbuf, norms, norms, m, idx2);

        // GAT layer 1: ft = feat @ W1^T  (2048 x 6400 x 6400)
        k_gemm_nt_wmma<<<dim3(f / 64, NN / 64), 256, 0, stream>>>(
            feat, W1, ft, NN, f, f);
        k_elr<<<NN, 256, 0, stream>>>(ft, al1, ar1, el, er, f);
        k_agg<<<m, 256, 0, stream>>>(ft, el, er, idx2, idx1, agg, m, f);
        k_bias_relu<<<(unsigned)((NF + 255) / 256), 256, 0, stream>>>(
            agg, b1, feat, m, NN, f);

        // GAT layer 2
        k_gemm_nt_wmma<<<dim3(f / 64, NN / 64), 256, 0, stream>>>(
            feat, W2, ft, NN, f, f);
        k_elr<<<NN, 256, 0, stream>>>(ft, al2, ar2, el, er, f);
        k_agg<<<m, 256, 0, stream>>>(ft, el, er, idx2, idx1, agg, m, f);

        // out[b][col][mm] = agg[mm][col] + b2[col]  (transpose + bias)
        k_transpose_bias<<<dim3(m / 32, f / 32), 256, 0, stream>>>(
            agg, outb, b2, m, f);
    }
}